// GraphSAGEBatch_87247965651354
// MI455X (gfx1250) — compile-verified
//
#include <hip/hip_runtime.h>

typedef float v2f __attribute__((ext_vector_type(2)));
typedef float v8f __attribute__((ext_vector_type(8)));

#define HID 128

// ---------------- zero fill (ws is poisoned 0xAA; must re-zero every call) ---
__global__ __launch_bounds__(256) void zero_f32(float* __restrict__ p, int n) {
  int i = blockIdx.x * blockDim.x + threadIdx.x;
  if (i < n) p[i] = 0.0f;
}

__device__ __forceinline__ void atom_add_f32(float* p, float v) {
  // native global_atomic_add_f32 path on gfx1250 (agg lives in L2: 63MB << 192MB)
  unsafeAtomicAdd(p, v);
}

// ---------------- edge scatter: one wave32 per edge ------------------------
// Each lane loads a float4 (32 lanes * 16B = 512B = one 128-f32 row) from the
// source row and atomically accumulates into agg[dst]; lane 0 bumps degree.
__global__ __launch_bounds__(256) void sage_scatter(
    const float* __restrict__ Hsrc,
    const int*   __restrict__ src,
    const int*   __restrict__ dst,
    float* __restrict__ Agg,
    float* __restrict__ Deg,
    int E)
{
  const int lane = threadIdx.x & 31;
  const int e = blockIdx.x * (blockDim.x >> 5) + (threadIdx.x >> 5);
  if (e >= E) return;
  const int s = src[e];
  const int d = dst[e];
  const float4 v = *(const float4*)(Hsrc + (size_t)s * HID + lane * 4);
  float* ap = Agg + (size_t)d * HID + lane * 4;
  atom_add_f32(ap + 0, v.x);
  atom_add_f32(ap + 1, v.y);
  atom_add_f32(ap + 2, v.z);
  atom_add_f32(ap + 3, v.w);
  if (lane == 0) atom_add_f32(Deg + d, 1.0f);
}

// ---------------- fused dual-GEMM via V_WMMA_F32_16X16X4_F32 ----------------
// Out[m0:m0+16, n0:n0+16] = act( Hsrc*Wself + (Agg/max(deg,1))*Wneigh + b )
// A 16x4 f32 layout: lanes 0-15 -> M=lane, K={k,k+1}; lanes 16-31 -> K={k+2,k+3}
// B 4x16 f32 layout: lanes 0-15 -> N=lane, K={k,k+1}; lanes 16-31 -> K={k+2,k+3}
// C/D 16x16:        vgpr r, lanes 0-15 -> (M=r, N=lane); lanes 16-31 -> (M=r+8)
// N and RELU are compile-time: no masking VALU on the N%16==0 layers, and all
// B addressing folds into immediate offsets.
template <int N, bool RELU>
__global__ __launch_bounds__(256) void sage_gemm_wmma(
    const float* __restrict__ Hsrc,
    const float* __restrict__ Agg,
    const float* __restrict__ Deg,
    const float* __restrict__ Wself,
    const float* __restrict__ Wneigh,
    const float* __restrict__ bias,
    float* __restrict__ Out)
{
  constexpr bool MASKED = (N % 16) != 0;

  const int lane = threadIdx.x & 31;
  const int wave = threadIdx.x >> 5;
  const int half = lane >> 4;        // which K-pair this lane supplies
  const int l15  = lane & 15;
  const int m0 = blockIdx.x << 4;    // M is always a multiple of 16
  const int n0 = wave << 4;

  const int mA = m0 + l15;
  const int nB = n0 + l15;
  int   nBc   = nB;
  float nMask = 1.0f;
  if constexpr (MASKED) {
    nBc   = (nB < N) ? nB : (N - 1);   // clamp tail column
    nMask = (nB < N) ? 1.0f : 0.0f;    // VALU mask keeps EXEC full for WMMA
  }

  const float invd = 1.0f / fmaxf(Deg[mA], 1.0f);
  const float* hrow = Hsrc + (size_t)mA * HID;
  const float* arow = Agg  + (size_t)mA * HID;

  v8f c = {0.f, 0.f, 0.f, 0.f, 0.f, 0.f, 0.f, 0.f};

  #pragma unroll 4
  for (int k0 = 0; k0 < HID; k0 += 4) {
    const int kk = k0 + 2 * half;
    v2f as = *(const v2f*)(hrow + kk);               // self features
    v2f an = *(const v2f*)(arow + kk);               // aggregated features
    an.x *= invd; an.y *= invd;                      // fused mean
    v2f bs, bn;
    bs.x = Wself [(size_t)(kk    ) * N + nBc];
    bs.y = Wself [(size_t)(kk + 1) * N + nBc];
    bn.x = Wneigh[(size_t)(kk    ) * N + nBc];
    bn.y = Wneigh[(size_t)(kk + 1) * N + nBc];
    if constexpr (MASKED) {
      bs.x *= nMask; bs.y *= nMask;
      bn.x *= nMask; bn.y *= nMask;
    }
    c = __builtin_amdgcn_wmma_f32_16x16x4_f32(false, as, false, bs, (short)0, c, false, false);
    c = __builtin_amdgcn_wmma_f32_16x16x4_f32(false, an, false, bn, (short)0, c, false, false);
  }

  if (!MASKED || nB < N) {
    const float bv = bias[nB];
    #pragma unroll
    for (int r = 0; r < 8; ++r) {
      const int row = m0 + r + 8 * half;
      float v = c[r] + bv;
      if constexpr (RELU) v = fmaxf(v, 0.0f);
      Out[(size_t)row * N + nB] = v;
    }
  }
}

// ---------------------------------------------------------------------------
extern "C" void kernel_launch(void* const* d_in, const int* in_sizes, int n_in,
                              void* d_out, int out_size, void* d_ws, size_t ws_size,
                              hipStream_t stream)
{
  (void)in_sizes; (void)n_in; (void)out_size; (void)ws_size;

  const float* x     = (const float*)d_in[0];
  const float* Ws[3] = {(const float*)d_in[1], (const float*)d_in[4], (const float*)d_in[7]};
  const float* Wn[3] = {(const float*)d_in[2], (const float*)d_in[5], (const float*)d_in[8]};
  const float* bb[3] = {(const float*)d_in[3], (const float*)d_in[6], (const float*)d_in[9]};
  const int* srcI[3] = {(const int*)d_in[10], (const int*)d_in[12], (const int*)d_in[14]};
  const int* dstI[3] = {(const int*)d_in[11], (const int*)d_in[13], (const int*)d_in[15]};

  const int Marr[3] = {123904, 11264, 1024};
  const int Earr[3] = {1239040, 112640, 10240};

  char* ws = (char*)d_ws;
  float* agg = (float*)ws;                           // <= 63.5 MB, reused per layer
  float* deg = (float*)(ws + ((size_t)64  << 20));   // <= 0.5 MB, reused per layer
  float* h1  = (float*)(ws + ((size_t)66  << 20));   // 63.5 MB, layer-0 output
  float* h2  = (float*)(ws + ((size_t)130 << 20));   // 5.8 MB, layer-1 output

  const float* hin[3]  = {x, h1, h2};
  float*       hout[3] = {h1, h2, (float*)d_out};

  for (int l = 0; l < 3; ++l) {
    const int M = Marr[l], E = Earr[l];
    const int nAgg = M * HID;
    zero_f32<<<(nAgg + 255) / 256, 256, 0, stream>>>(agg, nAgg);
    zero_f32<<<(M + 255) / 256, 256, 0, stream>>>(deg, M);
    sage_scatter<<<(E + 7) / 8, 256, 0, stream>>>(hin[l], srcI[l], dstI[l], agg, deg, E);
    if (l == 0) {
      sage_gemm_wmma<128, true><<<M / 16, 256, 0, stream>>>(
          hin[l], agg, deg, Ws[l], Wn[l], bb[l], hout[l]);
    } else if (l == 1) {
      sage_gemm_wmma<128, true><<<M / 16, 256, 0, stream>>>(
          hin[l], agg, deg, Ws[l], Wn[l], bb[l], hout[l]);
    } else {
      // N=47 -> 3 n-tiles -> 96 threads (3 waves)
      sage_gemm_wmma<47, false><<<M / 16, 96, 0, stream>>>(
          hin[l], agg, deg, Ws[l], Wn[l], bb[l], hout[l]);
    }
  }
}